// Attention_32409823216292
// MI455X (gfx1250) — compile-verified
//
#include <hip/hip_runtime.h>

// Problem dims from the reference: B=8, S=2048, D=768, fp32 everywhere.
#define BDIM 8
#define SDIM 2048
#define DDIM 768
#define D4   (DDIM / 4)         // 192 float4 per row
#define SCHUNKS 32
#define SPER (SDIM / SCHUNKS)   // 64 rows per chunk

typedef __attribute__((ext_vector_type(2))) float v2f;
typedef __attribute__((ext_vector_type(8))) float v8f;

// ---------------------------------------------------------------------------
// Stage 1: partial column-sums of v over S chunks (deterministic, no atomics).
// grid = (1, SCHUNKS, BDIM) = 256 blocks, block = 192 (one float4 column each).
// ---------------------------------------------------------------------------
__global__ void vsum_part_kernel(const float4* __restrict__ v4,
                                 float4* __restrict__ vpart4) {
  const int d4 = threadIdx.x;                        // 0..191
  const int c  = blockIdx.y;
  const int b  = blockIdx.z;
  const float4* base = v4 + ((size_t)b * SDIM + (size_t)c * SPER) * D4 + d4;
  float4 acc = {0.f, 0.f, 0.f, 0.f};
  for (int s = 0; s < SPER; ++s) {
    if (s + 16 < SPER)
      __builtin_prefetch(base + (size_t)(s + 16) * D4, 0, 0);  // global_prefetch_b8
    const float4 t = base[(size_t)s * D4];
    acc.x += t.x; acc.y += t.y; acc.z += t.z; acc.w += t.w;
  }
  vpart4[((size_t)b * SCHUNKS + c) * D4 + d4] = acc;
}

// ---------------------------------------------------------------------------
// Stage 2: reduce SCHUNKS partials -> clamped, 16-row zero-padded vsum.
// Rows 0..7 = clamp(sum_s v, +/-1e5)  (the reference's nan_to_num on attn),
// rows 8..15 = 0 so the WMMA A-tile loads are branch-free.
// grid covers 16*D4 float4 slots.
// ---------------------------------------------------------------------------
__global__ void vsum_reduce_kernel(const float4* __restrict__ vpart4,
                                   float4* __restrict__ vsum4) {
  const int i4 = blockIdx.x * blockDim.x + threadIdx.x;
  if (i4 >= 16 * D4) return;
  if (i4 < BDIM * D4) {
    const int b  = i4 / D4;
    const int d4 = i4 - b * D4;
    float4 acc = {0.f, 0.f, 0.f, 0.f};
#pragma unroll
    for (int c = 0; c < SCHUNKS; ++c) {
      const float4 t = vpart4[((size_t)b * SCHUNKS + c) * D4 + d4];
      acc.x += t.x; acc.y += t.y; acc.z += t.z; acc.w += t.w;
    }
    acc.x = fminf(fmaxf(acc.x, -1e5f), 1e5f);
    acc.y = fminf(fmaxf(acc.y, -1e5f), 1e5f);
    acc.z = fminf(fmaxf(acc.z, -1e5f), 1e5f);
    acc.w = fminf(fmaxf(acc.w, -1e5f), 1e5f);
    vsum4[i4] = acc;
  } else {
    const float4 z = {0.f, 0.f, 0.f, 0.f};
    vsum4[i4] = z;
  }
}

// ---------------------------------------------------------------------------
// FC GEMM with fp32 WMMA: hrow[b,j] = sum_d vsum[b,d] * fc_w[j,d] + fc_b[j]
// M=16 (8 real rows + 8 zero pad), N=768, K=768. One wave per 16-wide N tile.
// grid = 48 blocks, block = 32 (one wave).
//
// V_WMMA_F32_16X16X4_F32 lane layout (ISA 7.12.2):
//   A 16x4: lanes 0-15 -> M=lane, K={k0,k0+1}; lanes 16-31 -> M=lane-16, K={k0+2,k0+3}
//   B 4x16: lanes 0-15 -> N=lane, K={k0,k0+1}; lanes 16-31 -> N=lane-16, K={k0+2,k0+3}
//   C/D:    VGPR r, lanes 0-15 -> M=r, N=lane (rows 8..15 in lanes 16-31 = pad)
// ---------------------------------------------------------------------------
__global__ void fc_wmma_kernel(const float* __restrict__ vsum_pad,
                               const float* __restrict__ fc_w,
                               const float* __restrict__ fc_b,
                               float* __restrict__ hrow) {
  const int lane = threadIdx.x & 31;
  const int sub  = lane & 15;
  const int hlf  = lane >> 4;
  const int j    = blockIdx.x * 16 + sub;

  const float* arow = vsum_pad + (size_t)sub * DDIM + 2 * hlf;   // 8B-aligned
  const float* brow = fc_w + (size_t)j * DDIM + 2 * hlf;         // 8B-aligned

  v8f c = {};
  for (int k0 = 0; k0 < DDIM; k0 += 4) {
    const v2f a  = *(const v2f*)(arow + k0);
    const v2f bm = *(const v2f*)(brow + k0);
    c = __builtin_amdgcn_wmma_f32_16x16x4_f32(
        /*neg_a=*/false, a, /*neg_b=*/false, bm,
        /*c_mod=*/(short)0, c, /*reuse_a=*/false, /*reuse_b=*/false);
  }

  if (hlf == 0) {
    const float bias = fc_b[j];
#pragma unroll
    for (int r = 0; r < BDIM; ++r)     // rows M=0..7 live in lanes 0-15, VGPR r
      hrow[r * DDIM + j] = c[r] + bias;
  }
}

// ---------------------------------------------------------------------------
// Fused residual + LayerNorm. One wave32 per (b,s) row; 6 float4 per lane.
// grid = B*S/8 blocks, block = 256 (8 waves/block).
// ---------------------------------------------------------------------------
__global__ void resid_ln_kernel(const float4* __restrict__ q4,
                                const float4* __restrict__ hrow4,
                                const float4* __restrict__ g4,
                                const float4* __restrict__ bb4,
                                float4* __restrict__ out4) {
  const int lane = threadIdx.x & 31;
  const int wv   = threadIdx.x >> 5;
  const int row  = blockIdx.x * 8 + wv;          // 0 .. B*S-1
  const int b    = row / SDIM;
  const size_t base = (size_t)row * D4;

  float4 x[6];
  float sum = 0.f;
#pragma unroll
  for (int i = 0; i < 6; ++i) {
    const int d4 = lane + i * 32;
    const float4 qv = q4[base + d4];
    const float4 hv = hrow4[(size_t)b * D4 + d4];
    x[i].x = qv.x + hv.x; x[i].y = qv.y + hv.y;
    x[i].z = qv.z + hv.z; x[i].w = qv.w + hv.w;
    sum += x[i].x + x[i].y + x[i].z + x[i].w;
  }
#pragma unroll
  for (int off = 16; off > 0; off >>= 1) sum += __shfl_xor(sum, off, 32);
  const float mu = sum * (1.f / DDIM);

  float var = 0.f;
#pragma unroll
  for (int i = 0; i < 6; ++i) {
    float t;
    t = x[i].x - mu; var += t * t;
    t = x[i].y - mu; var += t * t;
    t = x[i].z - mu; var += t * t;
    t = x[i].w - mu; var += t * t;
  }
#pragma unroll
  for (int off = 16; off > 0; off >>= 1) var += __shfl_xor(var, off, 32);
  const float inv = rsqrtf(var * (1.f / DDIM) + 1e-5f);

#pragma unroll
  for (int i = 0; i < 6; ++i) {
    const int d4 = lane + i * 32;
    const float4 g = g4[d4];
    const float4 bb = bb4[d4];
    float4 o;
    o.x = (x[i].x - mu) * inv * g.x + bb.x;
    o.y = (x[i].y - mu) * inv * g.y + bb.y;
    o.z = (x[i].z - mu) * inv * g.z + bb.z;
    o.w = (x[i].w - mu) * inv * g.w + bb.w;
    out4[base + d4] = o;
  }
}

// ---------------------------------------------------------------------------
// Inputs (setup_inputs order): 0=q 1=k 2=v 3=mask 4=fc_w 5=fc_b 6=ln_g 7=ln_b
//                              8=index 9=typeAdd_Del
// ---------------------------------------------------------------------------
extern "C" void kernel_launch(void* const* d_in, const int* in_sizes, int n_in,
                              void* d_out, int out_size, void* d_ws, size_t ws_size,
                              hipStream_t stream) {
  const float4* q4   = (const float4*)d_in[0];
  const float4* v4   = (const float4*)d_in[2];
  const float*  fc_w = (const float*)d_in[4];
  const float*  fc_b = (const float*)d_in[5];
  const float4* g4   = (const float4*)d_in[6];
  const float4* bb4  = (const float4*)d_in[7];
  float4* out4 = (float4*)d_out;

  // Workspace layout (all 16B-aligned offsets)
  float4* vpart4 = (float4*)d_ws;                                  // B*SCHUNKS*D4
  float4* vsum4  = vpart4 + (size_t)BDIM * SCHUNKS * D4;           // 16*D4 (padded)
  float*  hrow   = (float*)(vsum4 + 16 * D4);                      // B*DDIM

  // 1) Deterministic 2-stage column sum of v (memory-bound: 50 MB read)
  vsum_part_kernel<<<dim3(1, SCHUNKS, BDIM), 192, 0, stream>>>(v4, vpart4);
  vsum_reduce_kernel<<<(16 * D4 + 255) / 256, 256, 0, stream>>>(vpart4, vsum4);

  // 2) Tiny FC GEMM via fp32 WMMA on the padded/clamped vsum
  fc_wmma_kernel<<<DDIM / 16, 32, 0, stream>>>((const float*)vsum4, fc_w, fc_b, hrow);

  // 3) Fused residual + LayerNorm over all B*S rows (100 MB read+write)
  resid_ln_kernel<<<(BDIM * SDIM) / 8, 256, 0, stream>>>(
      q4, (const float4*)hrow, g4, bb4, out4);
}